// VariantCoeLinear1d_83064667505021
// MI455X (gfx1250) — compile-verified
//
#include <hip/hip_runtime.h>
#include <stdint.h>

// 1-D local Lax-Friedrichs trajectory solver for MI455X (gfx1250).
// One workgroup per batch row; state double-buffered in LDS; trajectory
// streamed out with gfx1250 async LDS->global DMA (ASYNCcnt-tracked),
// overlapped with the next step's compute.

#ifndef __has_builtin
#define __has_builtin(x) 0
#endif

#define NN     4096
#define BATCHN 128
#define TPB    512
#define EPT    8        // cells per thread: 512*8 = 4096

// The gfx1250 async-LDS builtins take pointer-to-int32x4 operands
// (per hipcc diagnostic: "__vector_size__(4*sizeof(int)) int __device__ *").
typedef int v4i __attribute__((vector_size(16)));
typedef __attribute__((address_space(1))) v4i as1_v4i;   // global
typedef __attribute__((address_space(3))) v4i as3_v4i;   // LDS

#if __has_builtin(__builtin_amdgcn_global_load_async_to_lds_b128) && \
    __has_builtin(__builtin_amdgcn_global_store_async_from_lds_b128)
#define HAVE_ASYNC_LDS 1
#else
#define HAVE_ASYNC_LDS 0
#endif

__device__ __forceinline__ as1_v4i* gp16(const float* p) {
  // generic -> global: numerically identical 64-bit address
  return (as1_v4i*)(uintptr_t)p;
}
__device__ __forceinline__ as3_v4i* lp16(const float* p) {
  // generic LDS pointer: low 32 bits are the LDS offset (aperture in high bits)
  return (as3_v4i*)(uint32_t)(uintptr_t)p;
}

__device__ __forceinline__ void wait_async_0() {
#if __has_builtin(__builtin_amdgcn_s_wait_asynccnt)
  __builtin_amdgcn_s_wait_asynccnt(0);
#else
  asm volatile("s_wait_asynccnt 0" ::: "memory");
#endif
}
__device__ __forceinline__ void wait_async_le2() {
#if __has_builtin(__builtin_amdgcn_s_wait_asynccnt)
  __builtin_amdgcn_s_wait_asynccnt(2);
#else
  asm volatile("s_wait_asynccnt 2" ::: "memory");
#endif
}
__device__ __forceinline__ void wait_ds_0() {
  asm volatile("s_wait_dscnt 0" ::: "memory");
}

#if HAVE_ASYNC_LDS
__device__ __forceinline__ void async_load16(const float* g, const float* l) {
  __builtin_amdgcn_global_load_async_to_lds_b128(gp16(g), lp16(l), 0, 0);
}
__device__ __forceinline__ void async_store16(const float* g, const float* l) {
  __builtin_amdgcn_global_store_async_from_lds_b128(gp16(g), lp16(l), 0, 0);
}
#endif

// flux f(u) and |f'(u)| exactly as the reference (BETA = 0.1)
__device__ __forceinline__ float flux_f(float u) {
  const float C = 0.1f / 12.0f;
  float u2 = u * u;
  return 0.5f * u * (3.0f - u2) +
         C * u2 * (0.75f - 2.0f * u + 1.5f * u2 - 0.25f * u2 * u2);
}
__device__ __forceinline__ float aspeed(float u) {
  const float C = 0.1f / 12.0f;
  float u2 = u * u, u3 = u2 * u;
  float d = 1.5f - 1.5f * u2 +
            C * (1.5f * u - 6.0f * u2 + 6.0f * u3 - 1.5f * u2 * u3);
  return fabsf(d);
}

__global__ __launch_bounds__(TPB) void lf_traj_kernel(
    const float* __restrict__ init, const int* __restrict__ pstep,
    float* __restrict__ out) {
  __shared__ __align__(16) float ubuf[2][NN];   // 32 KB double-buffered state

  const int b     = blockIdx.x;
  const int tid   = threadIdx.x;
  const int i0    = tid * EPT;
  const int steps = *pstep;

  const float* gin = init + (size_t)b * NN;

  // ---- init row -> LDS (async DMA on gfx1250) ----
#if HAVE_ASYNC_LDS
  async_load16(gin + i0,     &ubuf[0][i0]);
  async_load16(gin + i0 + 4, &ubuf[0][i0 + 4]);
  wait_async_0();
#else
  *(float4*)&ubuf[0][i0]     = *(const float4*)(gin + i0);
  *(float4*)&ubuf[0][i0 + 4] = *(const float4*)(gin + i0 + 4);
#endif
  __syncthreads();

  // own 8 cells live in registers across all steps
  float uo[EPT];
  {
    float4 a4 = *(const float4*)&ubuf[0][i0];
    float4 b4 = *(const float4*)&ubuf[0][i0 + 4];
    uo[0] = a4.x; uo[1] = a4.y; uo[2] = a4.z; uo[3] = a4.w;
    uo[4] = b4.x; uo[5] = b4.y; uo[6] = b4.z; uo[7] = b4.w;
  }

  // trajectory[0] = init
  {
    float* g0 = out + (size_t)b * NN;
#if HAVE_ASYNC_LDS
    async_store16(g0 + i0,     &ubuf[0][i0]);
    async_store16(g0 + i0 + 4, &ubuf[0][i0 + 4]);
#else
    *(float4*)(g0 + i0)     = make_float4(uo[0], uo[1], uo[2], uo[3]);
    *(float4*)(g0 + i0 + 4) = make_float4(uo[4], uo[5], uo[6], uo[7]);
#endif
  }

  const float r = 0.5f;  // DT/DX = (0.75*DX/1.5)/DX exactly

  for (int t = 1; t < steps; ++t) {
    const int cur = (t - 1) & 1;
    const int nxt = t & 1;

    // halo cells from LDS (visible via barrier at end of previous iteration)
    float uu[EPT + 2];
    uu[0]       = (i0 > 0)        ? ubuf[cur][i0 - 1]   : 0.0f;
    uu[EPT + 1] = (i0 + EPT < NN) ? ubuf[cur][i0 + EPT] : 0.0f;
#pragma unroll
    for (int k = 0; k < EPT; ++k) uu[k + 1] = uo[k];

    float ff[EPT + 2], aa[EPT + 2];
#pragma unroll
    for (int k = 0; k < EPT + 2; ++k) {
      ff[k] = flux_f(uu[k]);
      aa[k] = aspeed(uu[k]);
    }

    // interfaces i0-1 .. i0+7
    float fh[EPT + 1];
#pragma unroll
    for (int k = 0; k < EPT + 1; ++k) {
      fh[k] = 0.5f * (ff[k] + ff[k + 1]) -
              0.5f * fmaxf(aa[k], aa[k + 1]) * (uu[k + 1] - uu[k]);
    }

    float un[EPT];
#pragma unroll
    for (int k = 0; k < EPT; ++k) un[k] = uu[k + 1] - r * (fh[k + 1] - fh[k]);
    if (i0 == 0)        un[0]       = un[1];        // u[0]   = interior[0]
    if (i0 + EPT == NN) un[EPT - 1] = un[EPT - 2];  // u[N-1] = interior[-1]

#if HAVE_ASYNC_LDS
    // this buffer was last DMA-read at step t-2; stores complete in order,
    // so <=2 outstanding means those are done
    wait_async_le2();
#endif
    *(float4*)&ubuf[nxt][i0]     = make_float4(un[0], un[1], un[2], un[3]);
    *(float4*)&ubuf[nxt][i0 + 4] = make_float4(un[4], un[5], un[6], un[7]);

    float* gr = out + ((size_t)t * BATCHN + (size_t)b) * NN;
#if HAVE_ASYNC_LDS
    wait_ds_0();  // LDS writes landed before the DMA reads them
    async_store16(gr + i0,     &ubuf[nxt][i0]);
    async_store16(gr + i0 + 4, &ubuf[nxt][i0 + 4]);
#else
    *(float4*)(gr + i0)     = make_float4(un[0], un[1], un[2], un[3]);
    *(float4*)(gr + i0 + 4) = make_float4(un[4], un[5], un[6], un[7]);
#endif

#pragma unroll
    for (int k = 0; k < EPT; ++k) uo[k] = un[k];

    __syncthreads();  // async stores keep draining while next step computes
  }
  // S_ENDPGM implicitly waits for all counters (incl. ASYNCcnt)
}

extern "C" void kernel_launch(void* const* d_in, const int* in_sizes, int n_in,
                              void* d_out, int out_size, void* d_ws, size_t ws_size,
                              hipStream_t stream) {
  (void)in_sizes; (void)n_in; (void)out_size; (void)d_ws; (void)ws_size;
  const float* init  = (const float*)d_in[0];
  const int*   pstep = (const int*)d_in[1];
  float*       out   = (float*)d_out;
  lf_traj_kernel<<<dim3(BATCHN), dim3(TPB), 0, stream>>>(init, pstep, out);
}